// DRR_7533372637276
// MI455X (gfx1250) — compile-verified
//
#include <hip/hip_runtime.h>
#include <hip/hip_bf16.h>

// ---------------- problem constants (from reference) ----------------
#define DRR_H      150
#define DRR_W      150
#define NPIX       (DRR_H * DRR_W)      // 22500 rays
#define NX         133
#define NY         512
#define NZ         512
#define NPLX       (NX + 1)             // 134 x-planes
#define NPLY       (NY + 1)             // 513
#define NPLZ       (NZ + 1)             // 513
#define LPR        8                    // lanes (chunks) per ray
#define BIGA       3.0e38f
#define TSTRIDE    4                    // padded target row stride (floats)

typedef __attribute__((ext_vector_type(2))) float v2f;
typedef __attribute__((ext_vector_type(8))) float v8f;

// ---------------------------------------------------------------------------
// Stage 1: targets[22500][3] = coefs[22500][2] @ [u;v](2x3) + center
// V_WMMA_F32_16X16X4_F32 per 16-row tile:
//   A (16x4) rows = [t_m, s_m, 1, 0]   B (4x16) rows = [u, v, center, 0]
// ---------------------------------------------------------------------------
__global__ __launch_bounds__(256) void drr_targets_wmma(
    const float* __restrict__ params, float* __restrict__ targets)
{
    const int NT = (NPIX + 15) / 16;          // 1407 tiles
    int wave = threadIdx.x >> 5;
    int lane = threadIdx.x & 31;
    int tile = blockIdx.x * 8 + wave;
    if (tile >= NT) return;                   // wave-uniform: EXEC all-ones at WMMA

    float sdr = params[0], th = params[1], ph = params[2], ga = params[3];
    float tx  = params[4], ty = params[5], tz = params[6];
    float ct = cosf(th), st = sinf(th);
    float cp = cosf(ph), sp = sinf(ph);
    float cg = cosf(ga), sg = sinf(ga);
    // R = Rz(th) Ry(ph) Rx(ga)
    float r00 = ct * cp, r10 = st * cp, r20 = -sp;                 // col 0
    float u0 = -st * cg + ct * sp * sg;                            // col 1 (u)
    float u1 =  ct * cg + st * sp * sg;
    float u2 =  cp * sg;
    float v0 =  st * sg + ct * sp * cg;                            // col 2 (v)
    float v1 = -ct * sg + st * sp * cg;
    float v2 =  cp * cg;
    float c0 = -sdr * r00 + tx;                                    // center
    float c1 = -sdr * r10 + ty;
    float c2 = -sdr * r20 + tz;

    bool hi  = lane >= 16;
    int  l15 = lane & 15;

    v2f a, b;
    if (!hi) {
        int m = tile * 16 + l15;
        if (m > NPIX - 1) m = NPIX - 1;       // tail rows: garbage ok, stores guarded
        int i = m / DRR_W, j = m % DRR_W;
        a.x = ((float)i - 74.5f) * 1.4f;      // t coef (K=0)
        a.y = ((float)j - 74.5f) * 1.4f;      // s coef (K=1)
        b.x = (l15 == 0) ? u0 : (l15 == 1) ? u1 : (l15 == 2) ? u2 : 0.f;  // B[0][n]
        b.y = (l15 == 0) ? v0 : (l15 == 1) ? v1 : (l15 == 2) ? v2 : 0.f;  // B[1][n]
    } else {
        a.x = 1.f;                            // K=2 selects center row
        a.y = 0.f;                            // K=3 zero row
        b.x = (l15 == 0) ? c0 : (l15 == 1) ? c1 : (l15 == 2) ? c2 : 0.f;  // B[2][n]
        b.y = 0.f;                                                         // B[3][n]
    }

    v8f acc = {};
    acc = __builtin_amdgcn_wmma_f32_16x16x4_f32(
        /*neg_a=*/false, a, /*neg_b=*/false, b,
        /*c_mod=*/(short)0, acc, /*reuse_a=*/false, /*reuse_b=*/false);

    int mbase = tile * 16 + (hi ? 8 : 0);
    bool full = (tile * 16 + 16) <= NPIX;     // wave-uniform
    if (full) {
        if (l15 < 3) {
#pragma unroll
            for (int r = 0; r < 8; ++r)
                targets[(mbase + r) * TSTRIDE + l15] = acc[r];
        }
    } else {
        if (l15 < 3) {
#pragma unroll
            for (int r = 0; r < 8; ++r) {
                int m = mbase + r;
                if (m < NPIX) targets[m * TSTRIDE + l15] = acc[r];
            }
        }
    }
}

// ---------------------------------------------------------------------------
// Stage 2: Siddon raycast, merged-crossing DDA with incremental flat address
// (Amanatides-Woo). 8 lanes per ray, each owns alpha chunk [ac, ac1).
// ---------------------------------------------------------------------------
#define AXIS_SETUP(sdd, src, spc, NP, nA, stpA, cnt)                          \
    {                                                                         \
        float q_ = (ac * (sdd) + (src)) / (spc);  /* well-conditioned */      \
        int i0_;                                                              \
        if ((sdd) > 0.f) {                                                    \
            i0_ = (int)floorf(q_) + 1;                                        \
            if (i0_ < 0) i0_ = 0;                                             \
            cnt = (NP) - i0_;                                                 \
        } else {                                                              \
            i0_ = (int)ceilf(q_) - 1;                                         \
            if (i0_ > (NP) - 1) i0_ = (NP) - 1;                               \
            cnt = i0_ + 1;                                                    \
        }                                                                     \
        stpA = fabsf((spc) / (sdd));                                          \
        nA = ((float)i0_ * (spc) - (src)) / (sdd);                            \
        if (cnt <= 0) nA = BIGA;                                              \
    }

__global__ __launch_bounds__(256) void drr_raycast(
    const float* __restrict__ params,  const float* __restrict__ volume,
    const float* __restrict__ spacing, const float* __restrict__ targets,
    float* __restrict__ out)
{
    int g     = blockIdx.x * blockDim.x + threadIdx.x;
    int ray   = g >> 3;
    int chunk = g & (LPR - 1);
    bool active = ray < NPIX;
    int  r      = active ? ray : NPIX - 1;   // clamp: redundant work, guarded store

    // source = sdr * R[:,0] + trans
    float sdr = params[0], th = params[1], ph = params[2];
    float ct = cosf(th), st = sinf(th), cp = cosf(ph), sp = sinf(ph);
    float sx = sdr * ct * cp + params[4];
    float sy = sdr * st * cp + params[5];
    float sz = -sdr * sp     + params[6];

    float sp0 = spacing[0], sp1 = spacing[1], sp2 = spacing[2];
    float L0 = (float)NX * sp0, L1 = (float)NY * sp1, L2 = (float)NZ * sp2;

    float dx = targets[r * TSTRIDE + 0] - sx + 1e-17f;
    float dy = targets[r * TSTRIDE + 1] - sy + 1e-17f;
    float dz = targets[r * TSTRIDE + 2] - sz + 1e-17f;
    float raylen = sqrtf(dx * dx + dy * dy + dz * dz);

    float a0x = -sx / dx, a1x = (L0 - sx) / dx;
    float a0y = -sy / dy, a1y = (L1 - sy) / dy;
    float a0z = -sz / dz, a1z = (L2 - sz) / dz;
    float amin = fmaxf(fmaxf(fminf(a0x, a1x), fminf(a0y, a1y)), fminf(a0z, a1z));
    float amax = fminf(fminf(fmaxf(a0x, a1x), fmaxf(a0y, a1y)), fmaxf(a0z, a1z));

    float acc = 0.f;
    if (amax > amin) {
        float span = (amax - amin) * (1.0f / (float)LPR);
        float ac   = amin + (float)chunk * span;
        float ac1  = (chunk == LPR - 1) ? amax : (ac + span);

        float n0, n1, n2, st0, st1, st2;
        int   c0, c1, c2;
        AXIS_SETUP(dx, sx, sp0, NPLX, n0, st0, c0);
        AXIS_SETUP(dy, sy, sp1, NPLY, n1, st1, c1);
        AXIS_SETUP(dz, sz, sp2, NPLZ, n2, st2, c2);

        // initial voxel from the first segment's midpoint (one-time divides)
        float an0 = fminf(fminf(n0, n1), n2);
        float mid = 0.5f * (ac + fminf(an0, ac1));
        int ix = (int)floorf((sx + mid * dx) / sp0);
        int iy = (int)floorf((sy + mid * dy) / sp1);
        int iz = (int)floorf((sz + mid * dz) / sp2);
        ix = ix < 0 ? 0 : (ix > NX - 1 ? NX - 1 : ix);
        iy = iy < 0 ? 0 : (iy > NY - 1 ? NY - 1 : iy);
        iz = iz < 0 ? 0 : (iz > NZ - 1 ? NZ - 1 : iz);

        // flat address into the dim0-flipped volume; per-axis address deltas
        int addr = (NX - 1 - ix) * (NY * NZ) + iy * NZ + iz;
        int dix = (dx > 0.f) ? 1 : -1;
        int diy = (dy > 0.f) ? 1 : -1;
        int diz = (dz > 0.f) ? 1 : -1;
        int adx = -dix * (NY * NZ);           // flip: ix+1 -> addr - NY*NZ
        int ady =  diy * NZ;
        int adz =  diz;

        float cur = ac;
        for (;;) {
            float an; int ax;
            if (n0 <= n1) { an = n0; ax = 0; } else { an = n1; ax = 1; }
            if (n2 < an)  { an = n2; ax = 2; }
            float ae  = fminf(an, ac1);
            float seg = fmaxf(ae - cur, 0.f);
            acc = fmaf(volume[addr], seg, acc);
            if (!(an < ac1)) break;
            cur = an;
            if (ax == 0) {
                n0 += st0; if (--c0 == 0) n0 = BIGA;
                int t = ix + dix; if ((unsigned)t <= NX - 1) { ix = t; addr += adx; }
            } else if (ax == 1) {
                n1 += st1; if (--c1 == 0) n1 = BIGA;
                int t = iy + diy; if ((unsigned)t <= NY - 1) { iy = t; addr += ady; }
            } else {
                n2 += st2; if (--c2 == 0) n2 = BIGA;
                int t = iz + diz; if ((unsigned)t <= NZ - 1) { iz = t; addr += adz; }
            }
        }
    }

    // sum the 8 chunk partials of each ray (xor masks 1,2,4 stay in-group)
    acc += __shfl_xor(acc, 1);
    acc += __shfl_xor(acc, 2);
    acc += __shfl_xor(acc, 4);

    if (active && chunk == 0) out[ray] = acc * raylen;
}

// ---------------------------------------------------------------------------
extern "C" void kernel_launch(void* const* d_in, const int* in_sizes, int n_in,
                              void* d_out, int out_size, void* d_ws, size_t ws_size,
                              hipStream_t stream)
{
    (void)in_sizes; (void)n_in; (void)out_size; (void)ws_size;
    const float* params  = (const float*)d_in[0];   // 7 floats
    const float* volume  = (const float*)d_in[1];   // 133*512*512 f32
    const float* spacing = (const float*)d_in[2];   // 3 floats
    float* out     = (float*)d_out;                 // 22500 f32
    float* targets = (float*)d_ws;                  // 22500*4 f32 scratch (360 KB)

    const int NT = (NPIX + 15) / 16;                // 1407 tiles
    dim3 grid1((NT + 7) / 8);                       // 8 waves/block
    drr_targets_wmma<<<grid1, 256, 0, stream>>>(params, targets);

    int threads = NPIX * LPR;                       // 180000
    dim3 grid2((threads + 255) / 256);              // 704 blocks
    drr_raycast<<<grid2, 256, 0, stream>>>(params, volume, spacing, targets, out);
}